// CNNGCN_7885559955672
// MI455X (gfx1250) — compile-verified
//
#include <hip/hip_runtime.h>
#include <hip/hip_bf16.h>

#define NN 50000
#define NE 800000
#define TLEN 518
#define KTAP 32
#define CDIM 83
#define KP1 84      // padded K for GEMM1
#define HID 64
#define OUTD 32

typedef __attribute__((ext_vector_type(2))) float v2f;
typedef __attribute__((ext_vector_type(8))) float v8f;

// ---------------------------------------------------------------------------
// Async global->LDS staging (gfx1250 ASYNCcnt path), with sync fallback so
// the file compiles on toolchains/passes without the builtin.
// Builtin signature (probed via diagnostics): first param is AS(1) int*
// (global src), second is AS(3) int* (LDS dst), then imm offset, imm cpol.
// ---------------------------------------------------------------------------
#if defined(__AMDGCN__) && __has_builtin(__builtin_amdgcn_global_load_async_to_lds_b32) && \
    __has_builtin(__builtin_amdgcn_s_wait_asynccnt)
#define USE_ASYNC_LDS 1
#else
#define USE_ASYNC_LDS 0
#endif

__device__ __forceinline__ void lds_async_copy_b32(float* lds_dst, const float* gsrc) {
#if USE_ASYNC_LDS
  // flat LDS address low 32 bits == LDS byte address (ISA 10.2 aperture rules)
  __builtin_amdgcn_global_load_async_to_lds_b32(
      (__attribute__((address_space(1))) int*)(unsigned long long)(__UINTPTR_TYPE__)gsrc,
      (__attribute__((address_space(3))) int*)(unsigned int)(__UINTPTR_TYPE__)lds_dst,
      0, 0);
#else
  *lds_dst = *gsrc;
#endif
}

__device__ __forceinline__ void lds_async_wait() {
#if USE_ASYNC_LDS
  __builtin_amdgcn_s_wait_asynccnt(0);
#endif
}

__device__ __forceinline__ void sched_fence() {
#if defined(__AMDGCN__) && __has_builtin(__builtin_amdgcn_sched_barrier)
  __builtin_amdgcn_sched_barrier(0);  // nothing may be reordered across this
#endif
}

// ---------------------------------------------------------------------------
// Fused dilated conv1 + conv2 (stride 2, dilation 2, VALID), one block/node.
// conv2 only consumes even-indexed conv1 outputs: y1e[t] = y1[2t], t<114.
// Row staged to LDS with async-to-LDS loads; output written to 84-wide
// zero-padded rows for WMMA GEMM1.
// ---------------------------------------------------------------------------
__global__ void conv_fused_kernel(const float* __restrict__ X,
                                  const float* __restrict__ w1,
                                  const float* __restrict__ b1,
                                  const float* __restrict__ w2,
                                  const float* __restrict__ b2,
                                  float* __restrict__ X2) {
  __shared__ float sx[TLEN];
  __shared__ float sy[114];
  __shared__ float sw1[KTAP];
  __shared__ float sw2[KTAP];
  const int node = blockIdx.x;
  const int tid = threadIdx.x;
  const float* xr = X + (size_t)node * TLEN;
  for (int i = tid; i < TLEN; i += blockDim.x) lds_async_copy_b32(&sx[i], &xr[i]);
  if (tid < KTAP) { sw1[tid] = w1[tid]; sw2[tid] = w2[tid]; }
  lds_async_wait();          // this wave's async LDS writes done
  __syncthreads();           // all waves' chunks visible
  const float bb1 = b1[0], bb2 = b2[0];
  for (int t = tid; t < 114; t += blockDim.x) {
    float acc = bb1;
#pragma unroll
    for (int k = 0; k < KTAP; ++k) acc += sw1[k] * sx[4 * t + 2 * k];
    sy[t] = fmaxf(acc, 0.0f);
  }
  __syncthreads();
  float* orow = X2 + (size_t)node * KP1;
  for (int o = tid; o < KP1; o += blockDim.x) {
    if (o < CDIM) {
      float acc = bb2;
#pragma unroll
      for (int k = 0; k < KTAP; ++k) acc += sw2[k] * sy[o + k];
      orow[o] = fmaxf(acc, 0.0f);
    } else {
      orow[o] = 0.0f;  // zero pad column 83
    }
  }
}

// Pad g1_w (83x64) -> (84x64) with a zero row.
__global__ void pad_w1_kernel(const float* __restrict__ W, float* __restrict__ Wp) {
  int i = blockIdx.x * blockDim.x + threadIdx.x;
  if (i >= KP1 * HID) return;
  int k = i / HID;
  Wp[i] = (k < CDIM) ? W[i] : 0.0f;
}

// ---------------------------------------------------------------------------
// FP32 WMMA GEMM: C[MxN] = A[MxK] * B[KxN]. One wave per 16x16 N-tile strip;
// the whole B panel for the strip (K x 16) is hoisted into registers once
// (v2f[K/4]). Per M-tile, ALL A fragments are staged into registers first
// (clause of b64 loads, staged s_wait_loadcnt), then the WMMA accumulation
// chain runs without per-instruction memory stalls. A sched_barrier pins the
// load/compute split; __launch_bounds__(32,1) removes occupancy pressure so
// all 21 fragments stay live.
// Lane layouts per CDNA5 ISA 7.12.2:
//   A 16x4 : lanes 0-15 hold row M=lane {K0,K1}; lanes 16-31 {K2,K3}
//   B 4x16 : VGPR v, lanes 0-15 -> K=v,  N=lane ; lanes 16-31 -> K=v+2
//   C 16x16: VGPR v, lanes 0-15 -> M=v,  N=lane ; lanes 16-31 -> M=v+8
// Requires K%4==0, N%16==0 (K=84/64, N=64/32 here). M-tile guard is
// wave-uniform so EXEC stays all-ones for WMMA.
// ---------------------------------------------------------------------------
template <int K, int MT>
__global__ void __launch_bounds__(32, 1)
wmma_gemm_f32(const float* __restrict__ A,
              const float* __restrict__ B,
              float* __restrict__ C,
              int M, int N) {
  constexpr int NC = K / 4;
  const int ntn = N >> 4;
  const int nt = blockIdx.x % ntn;
  const int mg = blockIdx.x / ntn;
  const int lane = threadIdx.x;
  const int half = lane >> 4;     // 0: K0/K1 side, 1: K2/K3 side
  const int lrow = lane & 15;
  const int n0 = nt << 4;

  v2f b[NC];
#pragma unroll
  for (int c = 0; c < NC; ++c) {
    const int kb = 4 * c + 2 * half;
    b[c].x = B[(size_t)kb * N + n0 + lrow];
    b[c].y = B[(size_t)(kb + 1) * N + n0 + lrow];
  }

  for (int mi = 0; mi < MT; ++mi) {
    const int m0 = (mg * MT + mi) << 4;
    if (m0 >= M) break;  // uniform across wave
    const float* arow = A + (size_t)(m0 + lrow) * K + 2 * half;
    __builtin_prefetch(arow + 16 * K, 0, 1);  // next M-tile row -> global_prefetch_b8

    // Stage all A fragments for this M-tile (clustered b64 loads).
    v2f a[NC];
#pragma unroll
    for (int c = 0; c < NC; ++c) {
      a[c].x = arow[4 * c];
      a[c].y = arow[4 * c + 1];
    }
    sched_fence();  // keep loads above, WMMA chain below

    // Uninterrupted WMMA accumulation chain.
    v8f acc = {};
#pragma unroll
    for (int c = 0; c < NC; ++c) {
      acc = __builtin_amdgcn_wmma_f32_16x16x4_f32(
          /*neg_a=*/false, a[c], /*neg_b=*/false, b[c],
          /*c_mod=*/(short)0, acc, /*reuse_a=*/false, /*reuse_b=*/false);
    }
#pragma unroll
    for (int v = 0; v < 8; ++v) {
      C[(size_t)(m0 + v + half * 8) * N + n0 + lrow] = acc[v];
    }
  }
}

// ---------------------------------------------------------------------------
// Degree / normalization
// ---------------------------------------------------------------------------
__global__ void deg_init_kernel(float* __restrict__ deg) {
  int i = blockIdx.x * blockDim.x + threadIdx.x;
  if (i < NN) deg[i] = 1.0f;  // self-loop weight
}

__global__ void deg_edges_kernel(const int* __restrict__ ei,
                                 const float* __restrict__ ew,
                                 float* __restrict__ deg) {
  int e = blockIdx.x * blockDim.x + threadIdx.x;
  if (e < NE) atomicAdd(&deg[ei[NE + e]], ew[e]);  // col = ei[1][e]
}

__global__ void dinv_kernel(const float* __restrict__ deg, float* __restrict__ dinv) {
  int i = blockIdx.x * blockDim.x + threadIdx.x;
  if (i < NN) {
    float d = deg[i];
    dinv[i] = (d > 0.0f) ? rsqrtf(d) : 0.0f;
  }
}

// out[i,f] = dinv[i]^2 * h[i,f] + bias[f]   (self-loop term + bias)
__global__ void agg_init_kernel(const float* __restrict__ h,
                                const float* __restrict__ dinv,
                                const float* __restrict__ bias,
                                float* __restrict__ out, int F) {
  int i = blockIdx.x * blockDim.x + threadIdx.x;
  if (i >= NN * F) return;
  int node = i / F;
  int f = i - node * F;
  float d = dinv[node];
  out[i] = d * d * h[i] + bias[f];
}

// one thread per (edge, feature): out[col,f] += dinv[row]*ew*dinv[col]*h[row,f]
__global__ void agg_edges_kernel(const int* __restrict__ ei,
                                 const float* __restrict__ ew,
                                 const float* __restrict__ dinv,
                                 const float* __restrict__ h,
                                 float* __restrict__ out, int F) {
  long long idx = (long long)blockIdx.x * blockDim.x + threadIdx.x;
  if (idx >= (long long)NE * F) return;
  int e = (int)(idx / F);
  int f = (int)(idx - (long long)e * F);
  int r = ei[e];
  int c = ei[NE + e];
  float norm = dinv[r] * ew[e] * dinv[c];
  atomicAdd(&out[(size_t)c * F + f], norm * h[(size_t)r * F + f]);
}

__global__ void relu_kernel(float* __restrict__ x, int n) {
  int i = blockIdx.x * blockDim.x + threadIdx.x;
  if (i < n) x[i] = fmaxf(x[i], 0.0f);
}

// One block per output feature; deterministic tree reduce over nodes.
__global__ void mean_kernel(const float* __restrict__ h, float* __restrict__ out) {
  __shared__ float ssum[256];
  const int f = blockIdx.x;
  const int tid = threadIdx.x;
  float acc = 0.0f;
  for (int i = tid; i < NN; i += blockDim.x) acc += h[(size_t)i * OUTD + f];
  ssum[tid] = acc;
  __syncthreads();
  for (int s = blockDim.x >> 1; s > 0; s >>= 1) {
    if (tid < s) ssum[tid] += ssum[tid + s];
    __syncthreads();
  }
  if (tid == 0) out[f] = ssum[0] * (1.0f / (float)NN);
}

// ---------------------------------------------------------------------------
extern "C" void kernel_launch(void* const* d_in, const int* in_sizes, int n_in,
                              void* d_out, int out_size, void* d_ws, size_t ws_size,
                              hipStream_t stream) {
  const float* node_features = (const float*)d_in[0];   // 50000 x 518
  const float* edge_attr     = (const float*)d_in[1];   // 800000
  const float* c1_w          = (const float*)d_in[2];   // 32
  const float* c1_b          = (const float*)d_in[3];   // 1
  const float* c2_w          = (const float*)d_in[4];   // 32
  const float* c2_b          = (const float*)d_in[5];   // 1
  const float* g1_w          = (const float*)d_in[6];   // 83 x 64
  const float* g1_b          = (const float*)d_in[7];   // 64
  const float* g2_w          = (const float*)d_in[8];   // 64 x 32
  const float* g2_b          = (const float*)d_in[9];   // 32
  const int*   edge_index    = (const int*)d_in[10];    // 2 x 800000
  float* out = (float*)d_out;                           // 32

  // Workspace carve-up (floats, 256B-aligned offsets)
  char* ws = (char*)d_ws;
  size_t off = 0;
  auto carve = [&](size_t nfloats) {
    float* p = (float*)(ws + off);
    off = (off + nfloats * sizeof(float) + 255) & ~(size_t)255;
    return p;
  };
  float* x2   = carve((size_t)NN * KP1);   // conv output, padded 84 cols
  float* w1p  = carve((size_t)KP1 * HID);  // padded g1_w
  float* h1   = carve((size_t)NN * HID);   // x2 @ w1p
  float* h1a  = carve((size_t)NN * HID);   // aggregated + relu
  float* h2   = carve((size_t)NN * OUTD);  // h1a @ g2_w
  float* h2a  = carve((size_t)NN * OUTD);  // aggregated
  float* deg  = carve((size_t)NN);
  float* dinv = carve((size_t)NN);
  (void)ws_size; (void)n_in; (void)in_sizes; (void)out_size;

  constexpr int MT = 4;                       // M-tiles per wave in the GEMMs
  const int mtiles = NN / 16;                 // 3125
  const int mgroups = (mtiles + MT - 1) / MT; // 782

  // Stage 1: fused convolutions -> padded features
  conv_fused_kernel<<<NN, 128, 0, stream>>>(node_features, c1_w, c1_b, c2_w, c2_b, x2);

  // Normalization terms (independent of conv)
  deg_init_kernel<<<(NN + 255) / 256, 256, 0, stream>>>(deg);
  deg_edges_kernel<<<(NE + 255) / 256, 256, 0, stream>>>(edge_index, edge_attr, deg);
  dinv_kernel<<<(NN + 255) / 256, 256, 0, stream>>>(deg, dinv);

  // Stage 2: GCN layer 1
  pad_w1_kernel<<<(KP1 * HID + 255) / 256, 256, 0, stream>>>(g1_w, w1p);
  wmma_gemm_f32<KP1, MT><<<mgroups * (HID / 16), 32, 0, stream>>>(x2, w1p, h1, NN, HID);
  agg_init_kernel<<<(NN * HID + 255) / 256, 256, 0, stream>>>(h1, dinv, g1_b, h1a, HID);
  {
    long long tot = (long long)NE * HID;
    agg_edges_kernel<<<(unsigned)((tot + 255) / 256), 256, 0, stream>>>(
        edge_index, edge_attr, dinv, h1, h1a, HID);
  }
  relu_kernel<<<(NN * HID + 255) / 256, 256, 0, stream>>>(h1a, NN * HID);

  // Stage 3: GCN layer 2
  wmma_gemm_f32<HID, MT><<<mgroups * (OUTD / 16), 32, 0, stream>>>(h1a, g2_w, h2, NN, OUTD);
  agg_init_kernel<<<(NN * OUTD + 255) / 256, 256, 0, stream>>>(h2, dinv, g2_b, h2a, OUTD);
  {
    long long tot = (long long)NE * OUTD;
    agg_edges_kernel<<<(unsigned)((tot + 255) / 256), 256, 0, stream>>>(
        edge_index, edge_attr, dinv, h2, h2a, OUTD);
  }

  // Stage 4: mean over nodes
  mean_kernel<<<OUTD, 256, 0, stream>>>(h2a, out);
}